// KVCache_70265664963052
// MI455X (gfx1250) — compile-verified
//
#include <hip/hip_runtime.h>
#include <hip/hip_bf16.h>

// SnapKV compression for MI455X (gfx1250, wave32).
// One workgroup (256 threads = 8 waves) per head. All selection state in LDS.
// priority = (1/16)*ones(1x16) * attn_tile via V_WMMA_F32_16X16X4_F32 (exact f32).
// top-k via radix-select on float bit patterns (keys cached in VGPRs) with
// parallel LDS scans; index-ordered compaction gives sorted indices for free.
// K/V gather: 1 wave per 512B row, float4 per lane -> global_{load,store}_b128.

typedef __attribute__((ext_vector_type(2))) float v2f;
typedef __attribute__((ext_vector_type(8))) float v8f;

#define S_LEN    8192
#define OBS      16
#define Q_LEN    64
#define D_HEAD   128
#define L_KEEP   2048
#define NTHREADS 256
#define NWAVES   8
#define PER_T    (S_LEN / NTHREADS)   // 32 elements per thread
#define PAD      2                    // pool radius guards on each side

__global__ __launch_bounds__(NTHREADS)
void snapkv_kernel(const float* __restrict__ attn,   // [H, Q, S]
                   const float* __restrict__ kval,   // [H, S, D]
                   const float* __restrict__ vval,   // [H, S, D]
                   float* __restrict__ kout,         // [H, L, D]
                   float* __restrict__ vout,         // [H, L, D]
                   float* __restrict__ ahout,        // [H, L]
                   int*   __restrict__ posout)       // [H, L]
{
    __shared__ float    prio_pad[S_LEN + 2 * PAD];  // 32KB+16B, zero guards
    __shared__ unsigned hist[256];        // radix histogram / suffix sums
    __shared__ unsigned tsum[NTHREADS];   // per-thread packed scan totals
    __shared__ int      keep_idx[L_KEEP]; // 8 KB: sorted kept indices
    __shared__ unsigned sh_prefix, sh_remaining;

    float* const prio = prio_pad + PAD;   // prio[s], s in [0, S_LEN)

    const int h    = blockIdx.x;
    const int tid  = threadIdx.x;
    const int lane = tid & 31;
    const int wave = tid >> 5;

    // zero guard cells (sum over zero-padded window == un-padded window sum)
    if (tid < 2 * PAD) {
        const int g = (tid < PAD) ? tid : (S_LEN + tid);  // 0,1, S+2,S+3
        prio_pad[g] = 0.0f;
    }

    // ---------------- Phase 1: priority via WMMA f32 16x16x4 ----------------
    // priority[s] = (1/16) * sum_{q in last 16} attn[h, q, s]
    // D = A(16x4, all 1/16) x B(4x16 attn slice) + C ; every row of D holds
    // the partial mean for 16 s-positions; accumulate 4 K-slices (16 queries).
    {
        const float* ap = attn + ((size_t)h * Q_LEN + (Q_LEN - OBS)) * S_LEN;
        v2f a; a.x = 1.0f / 16.0f; a.y = 1.0f / 16.0f;
        const int n   = lane & 15;   // s within tile (B cols, lanes 0-15 / 16-31)
        const int hlf = lane >> 4;   // which K row this half-wave supplies
        for (int chunk = wave; chunk < S_LEN / 16; chunk += NWAVES) {
            const int s0 = chunk * 16;
            v8f acc = {};
#pragma unroll
            for (int j = 0; j < 4; ++j) {
                v2f b;
                // B VGPR0: K = 4j + hlf ; B VGPR1: K = 4j + 2 + hlf
                b.x = ap[(size_t)(4 * j + hlf)     * S_LEN + s0 + n];
                b.y = ap[(size_t)(4 * j + 2 + hlf) * S_LEN + s0 + n];
                acc = __builtin_amdgcn_wmma_f32_16x16x4_f32(
                        false, a, false, b, (short)0, acc, false, false);
            }
            // D row M=0 lives in VGPR0 on lanes 0..15 (N = lane).
            if (lane < 16) prio[s0 + n] = acc[0];
        }
    }
    __syncthreads();

    // ---------------- Phase 1b: pooled keys -> registers (once) -------------
    // AvgPool1d(k=5, stride=1, pad=2, count_include_pad=False): zero guards
    // make the window SUM branch-free/statically addressed; only the divisor
    // depends on the clamped window size. Last OBS positions forced to 1.0.
    unsigned key[PER_T];
    {
        const int base = tid * PER_T;
        float w[PER_T + 2 * PAD];                 // prio[base-2 .. base+33]
#pragma unroll
        for (int j = 0; j < PER_T + 2 * PAD; ++j)
            w[j] = prio[base - PAD + j];          // static offsets, no clamps
#pragma unroll
        for (int e = 0; e < PER_T; ++e) {
            const int s = base + e;
            const float sum = w[e] + w[e + 1] + w[e + 2] + w[e + 3] + w[e + 4];
            const int lo = (s - 2 < 0) ? 0 : s - 2;
            const int hi = (s + 2 > S_LEN - 1) ? S_LEN - 1 : s + 2;
            float p = sum / (float)(hi - lo + 1);
            if (s >= S_LEN - OBS) p = 1.0f;
            key[e] = __float_as_uint(p);          // >=0 => uint order == float order
        }
    }

    // ---------------- Phase 2: radix-select the 2048-th largest -------------
    if (tid == 0) { sh_prefix = 0u; sh_remaining = L_KEEP; }
    __syncthreads();
    for (int shift = 24; shift >= 0; shift -= 8) {
        hist[tid] = 0u;
        __syncthreads();
        const unsigned pfx      = sh_prefix;
        const unsigned rem      = sh_remaining;
        const unsigned maskHigh = (unsigned)(0xFFFFFFFFull << (shift + 8));
#pragma unroll
        for (int e = 0; e < PER_T; ++e)
            if ((key[e] & maskHigh) == pfx)
                atomicAdd(&hist[(key[e] >> shift) & 0xFFu], 1u);
        __syncthreads();
        // parallel suffix-sum over the 256 bins (Hillis-Steele, 8 rounds)
        for (int d = 1; d < 256; d <<= 1) {
            const unsigned mine  = hist[tid];
            const unsigned other = (tid + d < 256) ? hist[tid + d] : 0u;
            __syncthreads();
            hist[tid] = mine + other;
            __syncthreads();
        }
        // unique winner: suffix[tid+1] < rem <= suffix[tid]
        const unsigned sHere  = hist[tid];
        const unsigned sAbove = (tid == 255) ? 0u : hist[tid + 1];
        if (sAbove < rem && rem <= sHere) {
            sh_prefix    = pfx | ((unsigned)tid << shift);
            sh_remaining = rem - sAbove;
        }
        __syncthreads();
    }
    const unsigned T        = sh_prefix;     // key of k-th largest pooled value
    const unsigned needTies = sh_remaining;  // how many ==T to keep (lowest idx)

    // ---------------- Phase 3: index-ordered scan -> sorted compaction ------
    // Packed counters: high16 = count(key > T), low16 = count(key == T).
    {
        unsigned total = 0;
#pragma unroll
        for (int e = 0; e < PER_T; ++e)
            total += (key[e] > T) ? 0x10000u : (key[e] == T ? 1u : 0u);
        tsum[tid] = total;
        __syncthreads();
        // parallel inclusive scan over 256 thread totals (8 rounds)
        for (int d = 1; d < 256; d <<= 1) {
            const unsigned mine  = tsum[tid];
            const unsigned other = (tid >= d) ? tsum[tid - d] : 0u;
            __syncthreads();
            tsum[tid] = mine + other;
            __syncthreads();
        }
        unsigned base = (tid == 0) ? 0u : tsum[tid - 1];   // exclusive
#pragma unroll
        for (int e = 0; e < PER_T; ++e) {
            const int s = tid * PER_T + e;
            const unsigned gtBefore = base >> 16;
            const unsigned eqBefore = base & 0xFFFFu;
            bool keep = false;
            if (key[e] > T)       keep = true;
            else if (key[e] == T) keep = (eqBefore < needTies);
            if (keep) {
                const unsigned eqKept = (eqBefore < needTies) ? eqBefore : needTies;
                const int opos = (int)(gtBefore + eqKept);
                keep_idx[opos]            = s;
                ahout[h * L_KEEP + opos]  = prio[s];
                posout[h * L_KEEP + opos] = s;
            }
            base += (key[e] > T) ? 0x10000u : (key[e] == T ? 1u : 0u);
        }
    }
    __syncthreads();

    // ---------------- Phase 4: cooperative K/V row gather -------------------
    // One wave per 512B row; one float4 (b128) per lane.
    {
        const float* kbase = kval + (size_t)h * S_LEN * D_HEAD;
        const float* vbase = vval + (size_t)h * S_LEN * D_HEAD;
        float* kob = kout + (size_t)h * L_KEEP * D_HEAD;
        float* vob = vout + (size_t)h * L_KEEP * D_HEAD;
        for (int i = wave; i < L_KEEP; i += NWAVES) {
            const int row = keep_idx[i];
            if (i + NWAVES < L_KEEP) {       // global_prefetch_b8 ahead
                const int nrow = keep_idx[i + NWAVES];
                __builtin_prefetch(kbase + (size_t)nrow * D_HEAD, 0, 0);
                __builtin_prefetch(vbase + (size_t)nrow * D_HEAD, 0, 0);
            }
            const float4* ks = (const float4*)(kbase + (size_t)row * D_HEAD);
            const float4* vs = (const float4*)(vbase + (size_t)row * D_HEAD);
            const float4 kv = ks[lane];
            const float4 vv = vs[lane];
            float4* kd = (float4*)(kob + (size_t)i * D_HEAD);
            float4* vd = (float4*)(vob + (size_t)i * D_HEAD);
            kd[lane] = kv;
            vd[lane] = vv;
        }
    }
}

extern "C" void kernel_launch(void* const* d_in, const int* in_sizes, int n_in,
                              void* d_out, int out_size, void* d_ws, size_t ws_size,
                              hipStream_t stream) {
    (void)in_sizes; (void)n_in; (void)out_size; (void)d_ws; (void)ws_size;
    const float* attn = (const float*)d_in[0];   // [1,32,64,8192] f32
    const float* kval = (const float*)d_in[1];   // [1,32,8192,128] f32
    const float* vval = (const float*)d_in[2];   // [1,32,8192,128] f32
    // d_in[3] = input_pos (unused by the outputs)

    float* outF = (float*)d_out;
    const size_t kvElems = (size_t)32 * L_KEEP * D_HEAD;   // 8388608
    float* kout  = outF;
    float* vout  = outF + kvElems;
    float* ahout = outF + 2 * kvElems;
    int*   posout = (int*)(outF + 2 * kvElems + (size_t)32 * L_KEEP);

    snapkv_kernel<<<dim3(32), dim3(NTHREADS), 0, stream>>>(
        attn, kval, vval, kout, vout, ahout, posout);
}